// Tree_RNNAgent_76098230550832
// MI455X (gfx1250) — compile-verified
//
#include <hip/hip_runtime.h>
#include <math.h>

// ---- problem dims ----
#define BS    512
#define NA    8
#define NN    (BS*NA)      // 4096
#define NE    8
#define NAL   7
#define FDIM  16
#define H     64
#define HEADS 4
#define HYPER 64
#define OUT_N 6
#define D_E   4356         // (16*64 + 64 + 1)*4
#define KBIG  2112         // 1024 (enemy Ubar) + 1024 (ally Ubar) + 16 + 16 + 16 (own) + 1 (bias) + 15 pad

typedef float v2f __attribute__((ext_vector_type(2)));
typedef float v4f __attribute__((ext_vector_type(4)));
typedef float v8f __attribute__((ext_vector_type(8)));

// ------------------------------------------------------------------
// Kernel 1: fold merge weights into GEMM-ready weight matrices.
// ------------------------------------------------------------------
__global__ void fold_weights(const float* __restrict__ he_w2, const float* __restrict__ he_b2,
                             const float* __restrict__ ha_w2, const float* __restrict__ ha_b2,
                             const float* __restrict__ wim,   const float* __restrict__ wom,
                             const float* __restrict__ fc1_w, const float* __restrict__ fc1_b,
                             const float* __restrict__ gw_ih, const float* __restrict__ gw_hh,
                             float* __restrict__ Wbig, float* __restrict__ WihT,
                             float* __restrict__ WhhT, float* __restrict__ W2mT,
                             float* __restrict__ wb,   float* __restrict__ bm,
                             float* __restrict__ t0)
{
  int tid = blockIdx.x * blockDim.x + threadIdx.x;
  // Wbig rows [0,2048): folded hypernet input-layer weights (enemy, then ally)
  if (tid < 2048*64) {
    int row = tid >> 6, h = tid & 63;
    int jf = row & 1023, j = jf >> 4, f = jf & 15;
    const float* src = (row < 1024) ? (he_w2 + (size_t)j * D_E)
                                    : (ha_w2 + (size_t)j * 4096);
    int c = f*256 + h*4;                           // col f*H*HEADS + h*HEADS + k
    float acc = 0.f;
    #pragma unroll
    for (int k = 0; k < HEADS; ++k) acc += src[c + k] * wim[k*64 + h];
    Wbig[(size_t)row*64 + h] = acc;
    return;
  }
  tid -= 2048*64;
  // Wbig rows [2048,2112): bias-features, fc1, constant, pad
  if (tid < 64*64) {
    int row = 2048 + (tid >> 6), h = tid & 63;
    float v = 0.f;
    if (row < 2064)      { int f = row-2048, c = f*256 + h*4;
                           for (int k=0;k<4;++k) v += he_b2[c+k]*wim[k*64+h]; }
    else if (row < 2080) { int f = row-2064, c = f*256 + h*4;
                           for (int k=0;k<4;++k) v += ha_b2[c+k]*wim[k*64+h]; }
    else if (row < 2096) { v = fc1_w[(row-2080)*64 + h]; }
    else if (row == 2096){ v = fc1_b[h]; }
    Wbig[(size_t)row*64 + h] = v;
    return;
  }
  tid -= 64*64;
  if (tid < 64*192) { int h = tid/192, o = tid%192; WihT[h*192+o] = gw_ih[o*64+h]; return; }
  tid -= 64*192;
  if (tid < 64*192) { int h = tid/192, o = tid%192; WhhT[h*192+o] = gw_hh[o*64+h]; return; }
  tid -= 64*192;
  if (tid < 64*64) { // W2mT[h][j] = sum_k wom[k]*he_w2[j, 4096 + h*4 + k]
    int h = tid >> 6, j = tid & 63; float acc = 0.f;
    for (int k=0;k<4;++k) acc += wom[k]*he_w2[(size_t)j*D_E + 4096 + h*4 + k];
    W2mT[h*64 + j] = acc; return;
  }
  tid -= 64*64;
  if (tid < 64) { float acc=0.f; for(int k=0;k<4;++k) acc += wom[k]*he_w2[(size_t)tid*D_E + 4352 + k];
                  wb[tid] = acc; return; }
  tid -= 64;
  if (tid < 64) { float acc=0.f; for(int k=0;k<4;++k) acc += wom[k]*he_b2[4096 + tid*4 + k];
                  bm[tid] = acc; return; }
  tid -= 64;
  if (tid == 0) { float acc=0.f; for(int k=0;k<4;++k) acc += wom[k]*he_b2[4352 + k]; t0[0] = acc; }
}

// ------------------------------------------------------------------
// Kernel 1b: pre-swizzle B[K][N] into per-lane WMMA fragment order.
// Bp[((kk*(N/64) + g)*32 + lane)*8 + s], s = tile(2b)|kpair(1b):
//   k   = 4*kk + 2*(lane>>4) + (s&1)
//   col = g*64 + (s>>1)*16 + (lane&15)
// One wave K-step then reads its 4 B fragments as two b128 loads.
// Requires N % 64 == 0, K % 4 == 0.
// ------------------------------------------------------------------
__global__ void pack_B(const float* __restrict__ B, float* __restrict__ Bp, int K, int N)
{
  int o = blockIdx.x * blockDim.x + threadIdx.x;
  if (o >= K * N) return;
  int s    = o & 7;
  int lane = (o >> 3) & 31;
  int rest = o >> 8;                 // kk*(N/64) + g
  int ng   = N >> 6;
  int g    = rest % ng;
  int kk   = rest / ng;
  int k    = 4*kk + 2*(lane >> 4) + (s & 1);
  int col  = g*64 + (s >> 1)*16 + (lane & 15);
  Bp[o] = B[(size_t)k * N + col];
}

// ------------------------------------------------------------------
// Kernel 2: per-agent entity pass. Computes first hyper layer r = relu(W1 f + b)
// for every enemy/ally, accumulates Ubar = sum_entities outer(r, f) into the
// unified A matrix, and fills the bias/own/const tail columns.
// One block per n (64 threads = 2 waves).
// ------------------------------------------------------------------
__global__ void build_A(const float* __restrict__ own, const float* __restrict__ ef,
                        const float* __restrict__ af,
                        const float* __restrict__ he_w1, const float* __restrict__ he_b1,
                        const float* __restrict__ ha_w1, const float* __restrict__ ha_b1,
                        float* __restrict__ Abig, float* __restrict__ r_e)
{
  const int n = blockIdx.x, j = threadIdx.x;            // j in [0,64)
  __shared__ float efs[NE][FDIM];
  __shared__ float afs[NAL][FDIM];
  for (int i = j; i < NE*FDIM;  i += 64) efs[i/FDIM][i%FDIM] = ef[(size_t)n*NE*FDIM + i];
  for (int i = j; i < NAL*FDIM; i += 64) afs[i/FDIM][i%FDIM] = af[(size_t)n*NAL*FDIM + i];
  __syncthreads();

  float* arow = Abig + (size_t)n * KBIG;
  float w1[FDIM];

  // ---- enemies ----
  #pragma unroll
  for (int f = 0; f < FDIM; ++f) w1[f] = he_w1[f*64 + j];
  float b1 = he_b1[j];
  float r[NE];
  #pragma unroll
  for (int e = 0; e < NE; ++e) {
    float acc = b1;
    #pragma unroll
    for (int f = 0; f < FDIM; ++f) acc += efs[e][f] * w1[f];
    acc = fmaxf(acc, 0.f);
    r[e] = acc;
    r_e[((size_t)n*NE + e)*64 + j] = acc;              // needed later for attack head
  }
  #pragma unroll
  for (int f = 0; f < FDIM; ++f) {
    float acc = 0.f;
    #pragma unroll
    for (int e = 0; e < NE; ++e) acc += r[e] * efs[e][f];
    arow[j*FDIM + f] = acc;                            // Ubar_e
  }

  // ---- allies ----
  #pragma unroll
  for (int f = 0; f < FDIM; ++f) w1[f] = ha_w1[f*64 + j];
  b1 = ha_b1[j];
  float ra[NAL];
  #pragma unroll
  for (int a = 0; a < NAL; ++a) {
    float acc = b1;
    #pragma unroll
    for (int f = 0; f < FDIM; ++f) acc += afs[a][f] * w1[f];
    ra[a] = fmaxf(acc, 0.f);
  }
  #pragma unroll
  for (int f = 0; f < FDIM; ++f) {
    float acc = 0.f;
    #pragma unroll
    for (int a = 0; a < NAL; ++a) acc += ra[a] * afs[a][f];
    arow[1024 + j*FDIM + f] = acc;                     // Ubar_a
  }

  // ---- tail columns ----
  if (j < 16) {
    float se = 0.f; for (int e=0;e<NE; ++e) se += efs[e][j];
    float sa = 0.f; for (int a=0;a<NAL;++a) sa += afs[a][j];
    arow[2048 + j] = se;
    arow[2064 + j] = sa;
    arow[2080 + j] = own[(size_t)n*FDIM + j];
  }
  if (j == 16) arow[2096] = 1.f;                       // bias column
  if (j >= 17 && j < 32) arow[2080 + j] = 0.f;         // pad cols 2097..2111
}

// ------------------------------------------------------------------
// f32 WMMA GEMM, register-blocked 16x64 per wave:
//   C[M,N] = A[M,K] @ B[K,N]  (B pre-swizzled by pack_B), optional relu.
// grid = (M/16, N/64), block = 32 (one wave).
// Per K-step: 1x global_load_b64 (A frag, reused by 4 WMMAs) +
//             2x global_load_b128 (4 B frags, lane-major packed) -> 4 WMMAs.
// A-frag per ISA: lanes 0-15 hold K{0,1}, lanes 16-31 hold K{2,3} (row = M).
// C tile: VGPR v -> row m0 + v + 8*half, col n0 + (lane&15).
// A rows stream from HBM once -> software prefetch ~1KB ahead
// (global_prefetch_b8 on gfx1250).
// ------------------------------------------------------------------
__device__ __forceinline__ void store_tile(float* crow, int N, int t, v8f acc, int do_relu)
{
  #pragma unroll
  for (int v = 0; v < 8; ++v) {
    float val = acc[v];
    if (do_relu) val = fmaxf(val, 0.f);
    crow[(size_t)v * N + t*16] = val;
  }
}

__global__ void gemm_wmma_f32(const float* __restrict__ A, const float* __restrict__ Bp,
                              float* __restrict__ C, int M, int N, int K, int do_relu)
{
  const int lane = threadIdx.x;          // 0..31, wave32
  const int m0   = blockIdx.x * 16;
  const int g    = blockIdx.y;           // 64-column group
  const int ng   = N >> 6;
  const int half = lane >> 4;            // 0 -> K pair {0,1}; 1 -> K pair {2,3}
  const int l15  = lane & 15;

  const float* arow = A + (size_t)(m0 + l15) * K + 2*half;   // 8B-aligned (K even)

  v8f acc0 = {}, acc1 = {}, acc2 = {}, acc3 = {};
  const int nkk = K >> 2;
  for (int kk = 0; kk < nkk; ++kk) {
    v2f a = *(const v2f*)(arow + 4*kk);
    __builtin_prefetch(arow + 4*kk + 256, 0, 3);             // stream A ahead
    const float* b = Bp + (((size_t)kk * ng + g) * 32 + lane) * 8;
    v4f b01 = *(const v4f*)(b);                              // tiles 0,1
    v4f b23 = *(const v4f*)(b + 4);                          // tiles 2,3
    v2f bA = __builtin_shufflevector(b01, b01, 0, 1);
    v2f bB = __builtin_shufflevector(b01, b01, 2, 3);
    v2f bC = __builtin_shufflevector(b23, b23, 0, 1);
    v2f bD = __builtin_shufflevector(b23, b23, 2, 3);
    acc0 = __builtin_amdgcn_wmma_f32_16x16x4_f32(false, a, false, bA, (short)0, acc0, false, false);
    acc1 = __builtin_amdgcn_wmma_f32_16x16x4_f32(false, a, false, bB, (short)0, acc1, false, false);
    acc2 = __builtin_amdgcn_wmma_f32_16x16x4_f32(false, a, false, bC, (short)0, acc2, false, false);
    acc3 = __builtin_amdgcn_wmma_f32_16x16x4_f32(false, a, false, bD, (short)0, acc3, false, false);
  }

  float* crow = C + (size_t)(m0 + 8*half) * N + g*64 + l15;
  store_tile(crow, N, 0, acc0, do_relu);
  store_tile(crow, N, 1, acc1, do_relu);
  store_tile(crow, N, 2, acc2, do_relu);
  store_tile(crow, N, 3, acc3, do_relu);
}

// ------------------------------------------------------------------
// Kernel: GRU gates (torch GRUCell semantics), elementwise over (n, h).
// ------------------------------------------------------------------
__global__ void gru_gates(const float* __restrict__ gi, const float* __restrict__ gh,
                          const float* __restrict__ b_ih, const float* __restrict__ b_hh,
                          const float* __restrict__ h_prev, float* __restrict__ hh)
{
  int t = blockIdx.x * blockDim.x + threadIdx.x;
  if (t >= NN*H) return;
  int n = t >> 6, h = t & 63;
  const float* gin = gi + (size_t)n * 192;
  const float* ghn = gh + (size_t)n * 192;
  float ir = gin[h]       + b_ih[h];
  float iz = gin[64 + h]  + b_ih[64 + h];
  float ig = gin[128 + h] + b_ih[128 + h];
  float hr = ghn[h]       + b_hh[h];
  float hz = ghn[64 + h]  + b_hh[64 + h];
  float hg = ghn[128 + h] + b_hh[128 + h];
  float r = 1.f / (1.f + expf(-(ir + hr)));
  float z = 1.f / (1.f + expf(-(iz + hz)));
  float g = tanhf(ig + r * hg);
  hh[t] = (1.f - z) * g + z * h_prev[t];
}

// ------------------------------------------------------------------
// Kernel: output heads. q_normal = hh@fc2 + b; q_attack[n,e] = r_e . s' + t.
// One block per n, 64 threads.
// ------------------------------------------------------------------
__global__ void heads(const float* __restrict__ hh, const float* __restrict__ s_raw,
                      const float* __restrict__ wb, const float* __restrict__ bm,
                      const float* __restrict__ t0, const float* __restrict__ r_e,
                      const float* __restrict__ fc2_w, const float* __restrict__ fc2_b,
                      float* __restrict__ q)
{
  const int n = blockIdx.x, j = threadIdx.x;
  __shared__ float hs[H], ss[H];
  hs[j] = hh[(size_t)n*H + j];
  ss[j] = s_raw[(size_t)n*H + j] + wb[j];
  __syncthreads();
  if (j < OUT_N) {                                       // normal-action Q
    float acc = fc2_b[j];
    for (int h = 0; h < H; ++h) acc += hs[h] * fc2_w[h*OUT_N + j];
    q[(size_t)n*(OUT_N+NE) + j] = acc;
  } else if (j < OUT_N + NE) {                           // attack Q
    int e = j - OUT_N;
    float t = t0[0];
    for (int h = 0; h < H; ++h) t += hs[h] * bm[h];
    const float* rr = r_e + ((size_t)n*NE + e)*64;
    float acc = t;
    for (int h = 0; h < H; ++h) acc += rr[h] * ss[h];
    q[(size_t)n*(OUT_N+NE) + OUT_N + e] = acc;
  }
}

// ------------------------------------------------------------------
extern "C" void kernel_launch(void* const* d_in, const int* in_sizes, int n_in,
                              void* d_out, int out_size, void* d_ws, size_t ws_size,
                              hipStream_t stream) {
  const float* own   = (const float*)d_in[0];
  const float* ef    = (const float*)d_in[1];
  const float* af    = (const float*)d_in[2];
  const float* hid   = (const float*)d_in[3];
  const float* fc1_w = (const float*)d_in[4];
  const float* fc1_b = (const float*)d_in[5];
  const float* he_w1 = (const float*)d_in[6];
  const float* he_b1 = (const float*)d_in[7];
  const float* he_w2 = (const float*)d_in[8];
  const float* he_b2 = (const float*)d_in[9];
  const float* ha_w1 = (const float*)d_in[10];
  const float* ha_b1 = (const float*)d_in[11];
  const float* ha_w2 = (const float*)d_in[12];
  const float* ha_b2 = (const float*)d_in[13];
  const float* wim   = (const float*)d_in[14];
  const float* gw_ih = (const float*)d_in[15];
  const float* gb_ih = (const float*)d_in[16];
  const float* gw_hh = (const float*)d_in[17];
  const float* gb_hh = (const float*)d_in[18];
  const float* fc2_w = (const float*)d_in[19];
  const float* fc2_b = (const float*)d_in[20];
  const float* wom   = (const float*)d_in[21];

  // workspace layout (floats); every segment (except trailing t0) is a
  // multiple of 64 floats -> 256B alignment for v2f/v4f vector loads.
  float* ws    = (float*)d_ws;
  float* Abig  = ws;                           // 4096*2112
  float* Wbig  = Abig  + (size_t)NN*KBIG;      // 2112*64
  float* WbigP = Wbig  + (size_t)KBIG*64;      // 2112*64 (packed)
  float* r_e   = WbigP + (size_t)KBIG*64;      // 32768*64
  float* X     = r_e   + (size_t)NN*NE*64;     // 4096*64
  float* WihT  = X     + (size_t)NN*64;        // 64*192
  float* WihP  = WihT  + 64*192;               // 64*192 (packed)
  float* WhhT  = WihP  + 64*192;               // 64*192
  float* WhhP  = WhhT  + 64*192;               // 64*192 (packed)
  float* gi    = WhhP  + 64*192;               // 4096*192
  float* gh    = gi    + (size_t)NN*192;       // 4096*192
  float* W2mT  = gh    + (size_t)NN*192;       // 64*64
  float* W2mP  = W2mT  + 64*64;                // 64*64 (packed)
  float* wb    = W2mP  + 64*64;                // 64
  float* bm    = wb    + 64;                   // 64
  float* s_raw = bm    + 64;                   // 4096*64
  float* t0    = s_raw + (size_t)NN*64;        // 1

  float* q_out  = (float*)d_out;                   // (BS,NA,14) flat
  float* hh_out = q_out + (size_t)NN*(OUT_N+NE);   // (BS,NA,H) flat

  // 1) fold merge weights (163,969 work items)
  fold_weights<<<641, 256, 0, stream>>>(he_w2, he_b2, ha_w2, ha_b2, wim, wom,
                                        fc1_w, fc1_b, gw_ih, gw_hh,
                                        Wbig, WihT, WhhT, W2mT, wb, bm, t0);
  // 1b) pre-swizzle all B operands into WMMA fragment order
  pack_B<<<(KBIG*64 + 255)/256, 256, 0, stream>>>(Wbig, WbigP, KBIG, 64);
  pack_B<<<(64*192 + 255)/256, 256, 0, stream>>>(WihT, WihP, 64, 192);
  pack_B<<<(64*192 + 255)/256, 256, 0, stream>>>(WhhT, WhhP, 64, 192);
  pack_B<<<(64*64  + 255)/256, 256, 0, stream>>>(W2mT, W2mP, 64, 64);
  // 2) entity pass -> unified A matrix + saved enemy activations
  build_A<<<NN, 64, 0, stream>>>(own, ef, af, he_w1, he_b1, ha_w1, ha_b1, Abig, r_e);
  // 3) embedding: X = relu(Abig @ Wbig)   (4096 x 2112 x 64)
  gemm_wmma_f32<<<dim3(NN/16, 1), 32, 0, stream>>>(Abig, WbigP, X, NN, 64, KBIG, 1);
  // 4) gi = X @ WihT   (4096 x 64 x 192)
  gemm_wmma_f32<<<dim3(NN/16, 3), 32, 0, stream>>>(X, WihP, gi, NN, 192, 64, 0);
  // 5) gh = hidden @ WhhT
  gemm_wmma_f32<<<dim3(NN/16, 3), 32, 0, stream>>>(hid, WhhP, gh, NN, 192, 64, 0);
  // 6) GRU gates -> hh (written straight into output slot)
  gru_gates<<<(NN*H + 255)/256, 256, 0, stream>>>(gi, gh, gb_ih, gb_hh, hid, hh_out);
  // 7) attack projection: s_raw = hh @ W2mT   (4096 x 64 x 64)
  gemm_wmma_f32<<<dim3(NN/16, 1), 32, 0, stream>>>(hh_out, W2mP, s_raw, NN, 64, 64, 0);
  // 8) output heads -> q
  heads<<<NN, 64, 0, stream>>>(hh_out, s_raw, wb, bm, t0, r_e, fc2_w, fc2_b, q_out);
}